// LabelGAT_84387517431816
// MI455X (gfx1250) — compile-verified
//
#include <hip/hip_runtime.h>
#include <hip/hip_bf16.h>

// ---------------------------------------------------------------------------
// GATv2 forward for MI455X (gfx1250, wave32).
//   Stage 1: XL = X*Wl, XR = X*Wr  via V_WMMA_F32_16X16X4_F32 (exact fp32)
//   Stage 2: per-edge attention score + segment-max (encoded uint atomicMax)
//   Stage 3: exp(score - max) + segment-sum (atomicAdd f32)
//   Stage 4: out[dst] += alpha * XL[src]  (global_atomic_add_f32, L2-resident)
// ---------------------------------------------------------------------------

typedef __attribute__((ext_vector_type(2))) float v2f;
typedef __attribute__((ext_vector_type(8))) float v8f;

#define DIMS 128     // D = H*C
#define HEADS 2
#define CHAN 64
#define NSLOPE 0.2f

// ---- order-preserving float <-> uint encoding for atomic max -------------
__device__ __forceinline__ unsigned fenc(float f) {
    unsigned u = __float_as_uint(f);
    return (u & 0x80000000u) ? ~u : (u | 0x80000000u);
}
__device__ __forceinline__ float fdec(unsigned u) {
    u = (u & 0x80000000u) ? (u & 0x7FFFFFFFu) : ~u;
    return __uint_as_float(u);
}

// ---------------------------------------------------------------------------
// Kernel 1: fp32 WMMA GEMM. Computes a 16x16 tile of [XL | XR] per wave.
// gridDim.x = N/16 row tiles, gridDim.y = 4, 4 waves per block -> 16 col tiles
// (col tiles 0..7 -> Wl/XL, 8..15 -> Wr/XR). K = 128, 32 WMMA steps of K=4.
//
// A fragment (16x4 f32, ISA 7.12.2): lanes 0-15 = M rows, VGPR0 holds K=0 (lo
// half) / K=2 (hi half), VGPR1 holds K=1 / K=3. B fragment mirrored with lane
// = N column. C/D: VGPR i -> row (i + 8*half), lane%16 = N column.
// ---------------------------------------------------------------------------
__global__ __launch_bounds__(128)
void gatv2_gemm_wmma(const float* __restrict__ X,
                     const float* __restrict__ Wl,
                     const float* __restrict__ Wr,
                     float* __restrict__ XL,
                     float* __restrict__ XR) {
    const int lane = threadIdx.x & 31;
    const int wave = threadIdx.x >> 5;                  // 0..3
    const int ct   = blockIdx.y * 4 + wave;             // 0..15 column tile
    const int rowT = blockIdx.x;                        // row tile

    const float* __restrict__ W = (ct < 8) ? Wl : Wr;
    float*       __restrict__ O = (ct < 8) ? XL : XR;
    const int col0 = (ct & 7) * 16;

    const int half = lane >> 4;                         // 0 or 1
    const int l15  = lane & 15;

    const float* __restrict__ xrow = X + (size_t)(rowT * 16 + l15) * DIMS;

    v8f c = {};
#pragma unroll
    for (int k0 = 0; k0 < DIMS; k0 += 4) {
        const int ka = k0 + 2 * half;
        v2f a, b;
        a.x = xrow[ka + 0];
        a.y = xrow[ka + 1];
        b.x = W[(size_t)(ka + 0) * DIMS + col0 + l15];
        b.y = W[(size_t)(ka + 1) * DIMS + col0 + l15];
        c = __builtin_amdgcn_wmma_f32_16x16x4_f32(
                /*neg_a=*/false, a, /*neg_b=*/false, b,
                /*c_mod=*/(short)0, c, /*reuse_a=*/false, /*reuse_b=*/false);
    }

    float* __restrict__ orow = O + (size_t)(rowT * 16 + half * 8) * DIMS + col0 + l15;
#pragma unroll
    for (int i = 0; i < 8; ++i)
        orow[(size_t)i * DIMS] = c[i];
}

// ---------------------------------------------------------------------------
// Kernel: out[n][:] = bias[:]
// ---------------------------------------------------------------------------
__global__ void gatv2_init_out(float* __restrict__ out,
                               const float* __restrict__ bias, int total) {
    int i = blockIdx.x * blockDim.x + threadIdx.x;
    if (i < total) out[i] = bias[i & (DIMS - 1)];
}

// ---------------------------------------------------------------------------
// Kernel 2: one wave per edge. Lane L covers channels [4L, 4L+4); lanes 0-15
// are head 0, lanes 16-31 head 1. score[e][h] = att[h] . leaky(xl[s]+xr[d]).
// Segment max via encoded atomicMax.
// ---------------------------------------------------------------------------
__global__ __launch_bounds__(256)
void gatv2_edge_score(const long long* __restrict__ src,
                      const long long* __restrict__ dst,
                      const float* __restrict__ XL,
                      const float* __restrict__ XR,
                      const float* __restrict__ att,    // H*C = 128 floats
                      float* __restrict__ score,        // [T][H]
                      unsigned* __restrict__ menc,      // [N][H]
                      int E, int T) {
    const int e = (blockIdx.x * blockDim.x + threadIdx.x) >> 5;
    if (e >= T) return;
    const int lane = threadIdx.x & 31;

    int s, d;
    if (e < E) { s = (int)src[e]; d = (int)dst[e]; }
    else       { s = d = e - E; }

    const float4 a = *(const float4*)(XL + (size_t)s * DIMS + lane * 4);
    const float4 b = *(const float4*)(XR + (size_t)d * DIMS + lane * 4);
    const float4 w = *(const float4*)(att + lane * 4);

    float v0 = a.x + b.x, v1 = a.y + b.y, v2 = a.z + b.z, v3 = a.w + b.w;
    v0 = (v0 > 0.f) ? v0 : NSLOPE * v0;
    v1 = (v1 > 0.f) ? v1 : NSLOPE * v1;
    v2 = (v2 > 0.f) ? v2 : NSLOPE * v2;
    v3 = (v3 > 0.f) ? v3 : NSLOPE * v3;
    float p = w.x * v0 + w.y * v1 + w.z * v2 + w.w * v3;

    // reduce within each 16-lane half (per head)
    p += __shfl_xor(p, 8, 32);
    p += __shfl_xor(p, 4, 32);
    p += __shfl_xor(p, 2, 32);
    p += __shfl_xor(p, 1, 32);

    if ((lane & 15) == 0) {
        const int h = lane >> 4;
        score[(size_t)e * HEADS + h] = p;
        atomicMax(&menc[(size_t)d * HEADS + h], fenc(p));
    }
}

// ---------------------------------------------------------------------------
// Kernel 3: num = exp(score - m[dst]); den[dst] += num  (one thread per edge)
// ---------------------------------------------------------------------------
__global__ __launch_bounds__(256)
void gatv2_edge_norm(const long long* __restrict__ dst,
                     float* __restrict__ score,         // in: score, out: num
                     const unsigned* __restrict__ menc,
                     float* __restrict__ den,
                     int E, int T) {
    const int e = blockIdx.x * blockDim.x + threadIdx.x;
    if (e >= T) return;
    const int d = (e < E) ? (int)dst[e] : e - E;

    const float m0 = fdec(menc[(size_t)d * HEADS + 0]);
    const float m1 = fdec(menc[(size_t)d * HEADS + 1]);
    const float n0 = __expf(score[(size_t)e * HEADS + 0] - m0);
    const float n1 = __expf(score[(size_t)e * HEADS + 1] - m1);
    score[(size_t)e * HEADS + 0] = n0;
    score[(size_t)e * HEADS + 1] = n1;
    atomicAdd(&den[(size_t)d * HEADS + 0], n0);
    atomicAdd(&den[(size_t)d * HEADS + 1], n1);
}

// ---------------------------------------------------------------------------
// Kernel 4: out[dst] += (num/den[dst]) * XL[src]  (one wave per edge)
// ---------------------------------------------------------------------------
__global__ __launch_bounds__(256)
void gatv2_edge_aggr(const long long* __restrict__ src,
                     const long long* __restrict__ dst,
                     const float* __restrict__ XL,
                     const float* __restrict__ num,     // [T][H]
                     const float* __restrict__ den,     // [N][H]
                     float* __restrict__ out,           // [N][D]
                     int E, int T) {
    const int e = (blockIdx.x * blockDim.x + threadIdx.x) >> 5;
    if (e >= T) return;
    const int lane = threadIdx.x & 31;

    int s, d;
    if (e < E) { s = (int)src[e]; d = (int)dst[e]; }
    else       { s = d = e - E; }

    const int h = lane >> 4;
    const float alpha = num[(size_t)e * HEADS + h] / den[(size_t)d * HEADS + h];

    const float4 v = *(const float4*)(XL + (size_t)s * DIMS + lane * 4);
    float* o = out + (size_t)d * DIMS + lane * 4;
    atomicAdd(o + 0, alpha * v.x);
    atomicAdd(o + 1, alpha * v.y);
    atomicAdd(o + 2, alpha * v.z);
    atomicAdd(o + 3, alpha * v.w);
}

// ---------------------------------------------------------------------------
// Host launch
// ---------------------------------------------------------------------------
extern "C" void kernel_launch(void* const* d_in, const int* in_sizes, int n_in,
                              void* d_out, int out_size, void* d_ws, size_t ws_size,
                              hipStream_t stream) {
    const float*     x    = (const float*)d_in[0];       // [N, 128]
    const long long* ei   = (const long long*)d_in[1];   // [2, E] int64
    const float*     Wl   = (const float*)d_in[2];       // [128, 128]
    const float*     Wr   = (const float*)d_in[3];       // [128, 128]
    const float*     att  = (const float*)d_in[4];       // [2, 64]
    const float*     bias = (const float*)d_in[5];       // [128]
    float*           out  = (float*)d_out;               // [N, 128]

    const int N = in_sizes[0] / DIMS;
    const int E = in_sizes[1] / 2;
    const int T = E + N;

    const long long* src = ei;
    const long long* dst = ei + E;

    // workspace layout (all 16B-aligned: every region is a multiple of 128 floats
    // except score/menc/den which are multiples of 2*N / 2*T words)
    float*    XL    = (float*)d_ws;                       // N*128
    float*    XR    = XL + (size_t)N * DIMS;              // N*128
    float*    score = XR + (size_t)N * DIMS;              // T*2
    unsigned* menc  = (unsigned*)(score + (size_t)T * 2); // N*2
    float*    den   = (float*)(menc + (size_t)N * 2);     // N*2

    // zero segment-max (encoding 0 < every real float) and denominators
    hipMemsetAsync(menc, 0, (size_t)N * 2 * sizeof(unsigned) * 2, stream);

    // out = bias
    {
        const int total = N * DIMS;
        gatv2_init_out<<<(total + 255) / 256, 256, 0, stream>>>(out, bias, total);
    }

    // Stage 1: XL/XR GEMMs (fp32 WMMA)
    {
        dim3 grid(N / 16, 4);
        gatv2_gemm_wmma<<<grid, 128, 0, stream>>>(x, Wl, Wr, XL, XR);
    }

    // Stage 2: edge scores + segment max (1 wave / edge, 8 waves / block)
    {
        const int blocks = (T + 7) / 8;
        gatv2_edge_score<<<blocks, 256, 0, stream>>>(src, dst, XL, XR, att,
                                                     score, menc, E, T);
    }

    // Stage 3: exp + segment sum (1 thread / edge)
    {
        const int blocks = (T + 255) / 256;
        gatv2_edge_norm<<<blocks, 256, 0, stream>>>(dst, score, menc, den, E, T);
    }

    // Stage 4: weighted aggregation (1 wave / edge)
    {
        const int blocks = (T + 7) / 8;
        gatv2_edge_aggr<<<blocks, 256, 0, stream>>>(src, dst, XL, score, den,
                                                    out, E, T);
    }
}